// SIMPA_46420006535908
// MI455X (gfx1250) — compile-verified
//
#include <hip/hip_runtime.h>

// SIMPA forward: 6 sparse row-normalized propagations over two graphs,
// D=64 features, accumulated into out[N,128] = [feat_p | feat_n].
// All feature arrays are 25.6MB -> L2-resident (192MB). Edge-parallel push
// with native global_atomic_add_f32, b128 feature gathers, coalesced edge
// metadata loads broadcast via 16-wide shuffles (ds_bpermute), and
// global_prefetch_b8 on the streamed edge arrays.

#define FEAT 64          // feature dim (float)
#define FEAT4 16         // feature dim in float4
#define BATCH 16         // edges per 16-lane group batch

static inline unsigned cdiv(unsigned a, unsigned b) { return (a + b - 1) / b; }

// deg[row[e]] += w[e]  (placed first so the disasm snippet shows the atomic
// lowering: expect a single global_atomic_add_f32, not a CAS loop)
__global__ void k_deg_accum(const int* __restrict__ row, const float* __restrict__ w,
                            int E, float* __restrict__ deg) {
    int i = blockIdx.x * blockDim.x + threadIdx.x;
    if (i < E) unsafeAtomicAdd(&deg[row[i]], w[i]);
}

// Edge-parallel scatter: y[row[e]] += (w[e]*dinv[row[e]]) * x[col[e]]
// Each 16-lane group handles BATCH=16 edges:
//   - lane `sub` loads edge (base+sub) metadata coalesced, computes coef
//   - loop j=0..15: broadcast (r,c,coef) via __shfl(.,j,16); every lane
//     gathers one float4 of x[c] and atomically adds into y[r].
__global__ void __launch_bounds__(256)
k_prop(const int* __restrict__ row, const int* __restrict__ col,
       const float* __restrict__ ew, const float* __restrict__ dinv,
       int E, const float* __restrict__ x, float* __restrict__ y) {
    const int sub  = threadIdx.x & 15;            // lane within edge-group
    const int grp  = threadIdx.x >> 4;            // group within block
    const int gpb  = blockDim.x >> 4;             // groups per block
    const int gid0 = blockIdx.x * gpb + grp;
    const int step = gridDim.x * gpb * BATCH;     // edges per full sweep

    for (int base = gid0 * BATCH; base < E; base += step) {
        // --- coalesced metadata for 16 edges ---
        int ee = base + sub;
        int   r_s = 0, c_s = 0;
        float coef_s = 0.0f;
        if (ee < E) {
            r_s = row[ee];
            c_s = col[ee];
            coef_s = ew[ee] * dinv[r_s];
        }
        // prefetch next sweep's metadata cachelines (global_prefetch_b8)
        int np = base + step + sub;
        if (np < E) {
            __builtin_prefetch(&row[np], 0, 0);
            __builtin_prefetch(&col[np], 0, 0);
            __builtin_prefetch(&ew[np],  0, 0);
        }
        const int m = (E - base < BATCH) ? (E - base) : BATCH;
        for (int j = 0; j < m; ++j) {
            int   r    = __shfl(r_s, j, 16);      // ds_bpermute_b32 (wave32)
            int   c    = __shfl(c_s, j, 16);
            float coef = __shfl(coef_s, j, 16);
            const float4 xv = *(const float4*)(x + ((size_t)c << 6) + (sub << 2));
            float* yp = y + ((size_t)r << 6) + (sub << 2);
            unsafeAtomicAdd(yp + 0, coef * xv.x);
            unsafeAtomicAdd(yp + 1, coef * xv.y);
            unsafeAtomicAdd(yp + 2, coef * xv.z);
            unsafeAtomicAdd(yp + 3, coef * xv.w);
        }
    }
}

__global__ void k_fill(float* __restrict__ p, int n, float v) {
    int i = blockIdx.x * blockDim.x + threadIdx.x;
    if (i < n) p[i] = v;
}

// in-place: deg -> (deg != 0) ? 1/deg : 0
__global__ void k_deg_inv(float* __restrict__ deg, int n) {
    int i = blockIdx.x * blockDim.x + threadIdx.x;
    if (i < n) {
        float d = deg[i];
        deg[i] = (d != 0.0f) ? (1.0f / d) : 0.0f;
    }
}

// y = (0.5 * deg_inv[node]) * x   (self-loop term of positive propagation)
__global__ void k_self_init(float4* __restrict__ y, const float4* __restrict__ x,
                            const float* __restrict__ dinv, int n4 /* = N*16 */) {
    int i = blockIdx.x * blockDim.x + threadIdx.x;
    if (i < n4) {
        float s = 0.5f * dinv[i >> 4];
        float4 v = x[i];
        y[i] = make_float4(s * v.x, s * v.y, s * v.z, s * v.w);
    }
}

// out[node, off..off+63] += wvec[widx] * t[node, :]   (float4 vectorized)
__global__ void k_axpy_out(float4* __restrict__ out, const float4* __restrict__ t,
                           const float* __restrict__ wvec, int widx, int off4,
                           int n4 /* = N*16 */) {
    int i = blockIdx.x * blockDim.x + threadIdx.x;
    if (i < n4) {
        float w = wvec[widx];                  // uniform -> scalar load
        int node = i >> 4, q = i & 15;
        float4 v = t[i];
        float4* op = &out[node * 32 + off4 + q];
        float4 o = *op;
        o.x += w * v.x; o.y += w * v.y; o.z += w * v.z; o.w += w * v.w;
        *op = o;
    }
}

// out[node,0:64] = w_p[0]*x_p[node,:] ; out[node,64:128] = 0
__global__ void k_init_out(float4* __restrict__ out, const float4* __restrict__ xp,
                           const float* __restrict__ wp, int n32 /* = N*32 */) {
    int i = blockIdx.x * blockDim.x + threadIdx.x;
    if (i < n32) {
        int node = i >> 5, q = i & 31;
        if (q < 16) {
            float w = wp[0];
            float4 v = xp[node * 16 + q];
            out[i] = make_float4(w * v.x, w * v.y, w * v.z, w * v.w);
        } else {
            out[i] = make_float4(0.f, 0.f, 0.f, 0.f);
        }
    }
}

extern "C" void kernel_launch(void* const* d_in, const int* in_sizes, int n_in,
                              void* d_out, int out_size, void* d_ws, size_t ws_size,
                              hipStream_t stream) {
    const int EP = in_sizes[1];          // positive edge count
    const int EN = in_sizes[3];          // negative edge count
    const int N  = in_sizes[4] / FEAT;   // num nodes

    const int*   eip = (const int*)d_in[0];
    const float* ewp = (const float*)d_in[1];
    const int*   ein = (const int*)d_in[2];
    const float* ewn = (const float*)d_in[3];
    const float* xp  = (const float*)d_in[4];
    const float* xn  = (const float*)d_in[5];
    const float* wp  = (const float*)d_in[6];   // 3 scalars
    const float* wn  = (const float*)d_in[7];   // 3 scalars

    const int* rowp = eip;            // edge_index_p[0]
    const int* colp = eip + EP;       // edge_index_p[1]
    const int* rown = ein;
    const int* coln = ein + EN;

    // workspace: deg_p[N] | deg_n[N] | t0[N*64] | t1[N*64]   (~52 MB)
    float* ws   = (float*)d_ws;
    float* degp = ws;
    float* degn = ws + (size_t)N;
    float* t0   = ws + 2 * (size_t)N;
    float* t1   = t0 + (size_t)N * FEAT;

    float4*       out4 = (float4*)d_out;
    const float4* xp4  = (const float4*)xp;
    const float4* xn4  = (const float4*)xn;

    const int B = 256;
    const int n4  = N * FEAT4;   // float4 elements per feature buffer
    const int n32 = N * 32;      // float4 elements in out
    const int nf  = N * FEAT;    // floats per feature buffer
    const unsigned PROP_BLOCKS = 4096;

    // --- degree / normalization (deg_p starts at 0.5 = positive self-loop fill) ---
    k_fill<<<cdiv(N, B), B, 0, stream>>>(degp, N, 0.5f);
    k_fill<<<cdiv(N, B), B, 0, stream>>>(degn, N, 0.0f);
    k_deg_accum<<<cdiv(EP, B), B, 0, stream>>>(rowp, ewp, EP, degp);
    k_deg_accum<<<cdiv(EN, B), B, 0, stream>>>(rown, ewn, EN, degn);
    k_deg_inv<<<cdiv(N, B), B, 0, stream>>>(degp, N);
    k_deg_inv<<<cdiv(N, B), B, 0, stream>>>(degn, N);

    // --- feat_p init: out[:,0:64] = w_p0 * x_p ; out[:,64:128] = 0 ---
    k_init_out<<<cdiv(n32, B), B, 0, stream>>>(out4, xp4, wp, n32);

    // --- positive chain: t0 = prop_p(x_p); t1 = prop_p(t0) ---
    k_self_init<<<cdiv(n4, B), B, 0, stream>>>((float4*)t0, xp4, degp, n4);
    k_prop<<<PROP_BLOCKS, B, 0, stream>>>(rowp, colp, ewp, degp, EP, xp, t0);
    k_axpy_out<<<cdiv(n4, B), B, 0, stream>>>(out4, (const float4*)t0, wp, 1, 0, n4);

    k_self_init<<<cdiv(n4, B), B, 0, stream>>>((float4*)t1, (const float4*)t0, degp, n4);
    k_prop<<<PROP_BLOCKS, B, 0, stream>>>(rowp, colp, ewp, degp, EP, t0, t1);
    k_axpy_out<<<cdiv(n4, B), B, 0, stream>>>(out4, (const float4*)t1, wp, 2, 0, n4);

    // --- negative chain part 1: t0 = prop_n(x_n); t1 = prop_p(t0) ---
    k_fill<<<cdiv(nf, B), B, 0, stream>>>(t0, nf, 0.0f);
    k_prop<<<PROP_BLOCKS, B, 0, stream>>>(rown, coln, ewn, degn, EN, xn, t0);
    k_axpy_out<<<cdiv(n4, B), B, 0, stream>>>(out4, (const float4*)t0, wn, 0, 16, n4);

    k_self_init<<<cdiv(n4, B), B, 0, stream>>>((float4*)t1, (const float4*)t0, degp, n4);
    k_prop<<<PROP_BLOCKS, B, 0, stream>>>(rowp, colp, ewp, degp, EP, t0, t1);
    k_axpy_out<<<cdiv(n4, B), B, 0, stream>>>(out4, (const float4*)t1, wn, 1, 16, n4);

    // --- negative chain part 2: t0 = prop_p(x_n); t1 = prop_n(t0) ---
    k_self_init<<<cdiv(n4, B), B, 0, stream>>>((float4*)t0, xn4, degp, n4);
    k_prop<<<PROP_BLOCKS, B, 0, stream>>>(rowp, colp, ewp, degp, EP, xn, t0);
    k_fill<<<cdiv(nf, B), B, 0, stream>>>(t1, nf, 0.0f);
    k_prop<<<PROP_BLOCKS, B, 0, stream>>>(rown, coln, ewn, degn, EN, t0, t1);
    k_axpy_out<<<cdiv(n4, B), B, 0, stream>>>(out4, (const float4*)t1, wn, 2, 16, n4);
}